// MemModule_3693671874650
// MI455X (gfx1250) — compile-verified
//
#include <hip/hip_runtime.h>
#include <math.h>

typedef float v2f __attribute__((ext_vector_type(2)));
typedef float v4f __attribute__((ext_vector_type(4)));
typedef float v8f __attribute__((ext_vector_type(8)));

#define MEMD   2000
#define FEA    256
#define ROWS   32          // pixels (w values) per workgroup: full w-line, b/h fixed
#define SSTR   2004        // padded LDS stride for 32 x 2000 score matrix (bank-offsets halves)
#define XSTR   36          // padded LDS stride for X / Y staging (16B aligned rows)
#define XYLEN  (FEA * XSTR)          // 9216 floats, 36,864 B  -> LDS offset 0
#define LDSLEN (XYLEN + ROWS * SSTR) // + 64,128 floats -> 293,376 B total
#define LAMBDA 0.0025f
#define EPSV   1e-12f

__global__ __launch_bounds__(256) void memae_fused_kernel(
    const float* __restrict__ x,   // [16, 256, 32, 32]
    const float* __restrict__ W,   // [2000, 256]
    float* __restrict__ y,         // [16, 256, 32, 32]
    float* __restrict__ att)       // [16, 2000, 32, 32]
{
  // Single LDS allocation with manual layout: XY at byte offset 0 so every
  // GEMM1 A-fragment ds_load offset (< 36KB) folds into the 16-bit DS
  // immediate; Ss lives above it (accessed via per-tile base registers).
  __shared__ __attribute__((aligned(16))) float lds_all[LDSLEN];
  float* const XY = lds_all;          // X tile (col-major [c][w]) then Y tile
  float* const Ss = lds_all + XYLEN;  // 32 x 2000 scores -> att

  const int tid  = threadIdx.x;
  const int lane = tid & 31;
  const int wave = tid >> 5;
  const int rl   = lane & 15;   // M%16 for A-frag rows / N for B-frag cols
  const int half = lane >> 4;   // selects K pair {0,1} vs {2,3}

  const int n0 = blockIdx.x * ROWS;   // 512 blocks cover N = 16384
  const int b  = n0 >> 10;            // 1024 pixels per batch image
  const int h  = (n0 & 1023) >> 5;    // full w-line per block

  // ---------- stage X tile: x[b, c, h, 0..31] -> XY[c*XSTR + w] ----------
  {
    const int c = tid;
    const float* src = x + (((size_t)b * FEA + c) * 32 + h) * 32;
#pragma unroll
    for (int q = 0; q < 8; ++q) {
      v4f v = *(const v4f*)(src + 4 * q);
      *(v4f*)(&XY[c * XSTR + 4 * q]) = v;
    }
  }
  __syncthreads();

  // ---------- GEMM1: S[32,2000] = X[32,256] * W^T, f32 WMMA 16x16x4 ----------
  {
    // per-lane bases: element (c = 2*half, r = rl) and the r+16 sub-tile
    const float* xb0 = &XY[2 * half * XSTR + rl];
    const float* xb1 = xb0 + 16;
    for (int jt = wave; jt < MEMD / 16; jt += 8) {
      v8f acc0 = {};   // rows 0..15
      v8f acc1 = {};   // rows 16..31
      const float* wrow = W + (size_t)(jt * 16 + rl) * FEA + 2 * half;
#pragma unroll 4
      for (int kk = 0; kk < FEA / 4; ++kk) {
        const int kb = kk * 4;
        const v2f bf = *(const v2f*)(wrow + kb);   // B[k,n]: W[jt*16+rl, kb+2*half+{0,1}]
        v2f a0, a1;
        a0.x = xb0[kb * XSTR];          a0.y = xb0[kb * XSTR + XSTR];
        a1.x = xb1[kb * XSTR];          a1.y = xb1[kb * XSTR + XSTR];
        acc0 = __builtin_amdgcn_wmma_f32_16x16x4_f32(false, a0, false, bf, (short)0, acc0, false, false);
        acc1 = __builtin_amdgcn_wmma_f32_16x16x4_f32(false, a1, false, bf, (short)0, acc1, false, false);
      }
      const int m     = jt * 16 + rl;       // N index
      const int rbase = half * 8;           // M = v + 8*half (+16 for sub-tile 1)
#pragma unroll
      for (int v = 0; v < 8; ++v) {
        Ss[(rbase + v) * SSTR + m]      = acc0[v];
        Ss[(rbase + v + 16) * SSTR + m] = acc1[v];
      }
    }
  }
  __syncthreads();

  // ---------- softmax + hard-shrink-relu + L1 normalize (rows independent) ----------
  // wave w handles rows 4w..4w+3; lanes stride the 2000 columns; butterfly reductions.
  for (int rr = 0; rr < 4; ++rr) {
    const int r = wave * 4 + rr;
    float* srow = &Ss[r * SSTR];

    float mx = -INFINITY;
    for (int m = lane; m < MEMD; m += 32) mx = fmaxf(mx, srow[m]);
#pragma unroll
    for (int o = 16; o; o >>= 1) mx = fmaxf(mx, __shfl_xor(mx, o));

    float sum = 0.f;
    for (int m = lane; m < MEMD; m += 32) {
      const float e = __expf(srow[m] - mx);
      srow[m] = e;
      sum += e;
    }
#pragma unroll
    for (int o = 16; o; o >>= 1) sum += __shfl_xor(sum, o);
    const float inv = 1.0f / sum;

    float den = 0.f;
    for (int m = lane; m < MEMD; m += 32) {
      const float a = srow[m] * inv;
      const float t = a - LAMBDA;
      const float s = fmaxf(t, 0.f) * a / (fabsf(t) + EPSV);   // hard_shrink_relu
      srow[m] = s;
      den += s;
    }
#pragma unroll
    for (int o = 16; o; o >>= 1) den += __shfl_xor(den, o);
    const float dinv = 1.0f / fmaxf(den, EPSV);

    for (int m = lane; m < MEMD; m += 32) srow[m] *= dinv;
  }
  __syncthreads();

  // ---------- write att: att[b, m, h, 0..31] (contiguous 128B lines, NT stores) ----------
  {
    float* attbase = att + (size_t)b * MEMD * 1024 + h * 32;
    for (int m = tid; m < MEMD; m += 256) {
      float* dst = attbase + (size_t)m * 1024;
#pragma unroll
      for (int q = 0; q < 8; ++q) {
        v4f v;
        v.x = Ss[(4 * q + 0) * SSTR + m];
        v.y = Ss[(4 * q + 1) * SSTR + m];
        v.z = Ss[(4 * q + 2) * SSTR + m];
        v.w = Ss[(4 * q + 3) * SSTR + m];
        __builtin_nontemporal_store(v, (v4f*)(dst + 4 * q));
      }
    }
  }

  // ---------- GEMM2: Y[32,256] = att[32,2000] * W[2000,256] ----------
  for (int jc = wave; jc < FEA / 16; jc += 8) {
    v8f acc0 = {};
    v8f acc1 = {};
    const float* wcol  = W + jc * 16 + rl;              // W[k, jc*16+rl]
    const float* arow0 = &Ss[rl * SSTR + 2 * half];
    const float* arow1 = &Ss[(rl + 16) * SSTR + 2 * half];
#pragma unroll 2
    for (int kk = 0; kk < MEMD / 4; ++kk) {
      const int kb = kk * 4;
      v2f bf;
      bf.x = wcol[(size_t)(kb + 2 * half) * FEA];
      bf.y = wcol[(size_t)(kb + 2 * half + 1) * FEA];
      v2f a0, a1;
      a0.x = arow0[kb];  a0.y = arow0[kb + 1];
      a1.x = arow1[kb];  a1.y = arow1[kb + 1];
      acc0 = __builtin_amdgcn_wmma_f32_16x16x4_f32(false, a0, false, bf, (short)0, acc0, false, false);
      acc1 = __builtin_amdgcn_wmma_f32_16x16x4_f32(false, a1, false, bf, (short)0, acc1, false, false);
    }
    // transpose through LDS for contiguous y stores: XY[c*XSTR + w]
    const int c = jc * 16 + rl;
#pragma unroll
    for (int v = 0; v < 8; ++v) {
      XY[c * XSTR + half * 8 + v]      = acc0[v];
      XY[c * XSTR + half * 8 + v + 16] = acc1[v];
    }
  }
  __syncthreads();

  // ---------- write y: y[b, c, h, 0..31] (contiguous 128B lines, NT stores) ----------
  {
    const int c = tid;
    float* dst = y + (((size_t)b * FEA + c) * 32 + h) * 32;
#pragma unroll
    for (int q = 0; q < 8; ++q) {
      v4f v;
      v.x = XY[c * XSTR + 4 * q + 0];
      v.y = XY[c * XSTR + 4 * q + 1];
      v.z = XY[c * XSTR + 4 * q + 2];
      v.w = XY[c * XSTR + 4 * q + 3];
      __builtin_nontemporal_store(v, (v4f*)(dst + 4 * q));
    }
  }
}

extern "C" void kernel_launch(void* const* d_in, const int* in_sizes, int n_in,
                              void* d_out, int out_size, void* d_ws, size_t ws_size,
                              hipStream_t stream) {
  (void)in_sizes; (void)n_in; (void)d_ws; (void)ws_size; (void)out_size;
  const float* x = (const float*)d_in[0];   // [16, 256, 32, 32]
  const float* w = (const float*)d_in[1];   // [2000, 256]
  float* y   = (float*)d_out;                         // first output, 16*256*32*32
  float* att = y + (size_t)16 * 256 * 32 * 32;        // second output, 16*2000*32*32
  // N = 16384 pixels / 32 per block = 512 blocks, 256 threads (8 wave32 waves)
  memae_fused_kernel<<<512, 256, 0, stream>>>(x, w, y, att);
}